// ModelLoss_5970004541458
// MI455X (gfx1250) — compile-verified
//
#include <hip/hip_runtime.h>
#include <math.h>

// Problem constants (match reference)
#define B_    32
#define P_    8732
#define C_    81
#define K_    16
#define PPAD  16384          // next pow2 >= P_, one LDS-sortable row (64KB)

// Workspace layout (32-bit words)
#define OFF_CENEG 0                         // float  [B_ * PPAD]
#define OFF_NPOS  (B_ * PPAD)               // int    [B_]
#define OFF_PFO   (OFF_NPOS + B_)           // int    [B_ * K_]  prior_for_obj
#define OFF_ACC   (OFF_PFO + B_ * K_)       // float  [3]: 0=l1_sum 1=ce_pos_sum 2=hard_sum
#define WS_WORDS  (OFF_ACC + 3)

#if __has_builtin(__builtin_amdgcn_global_load_async_to_lds_b128)
#define HAVE_ASYNC_LDS 1
typedef int v4i_async __attribute__((vector_size(16)));
typedef __attribute__((address_space(1))) v4i_async* gv4i_ptr;
typedef __attribute__((address_space(3))) v4i_async* lv4i_ptr;
#else
#define HAVE_ASYNC_LDS 0
#endif

// ---------------------------------------------------------------- zero init
__global__ void k_zero(unsigned* __restrict__ w, int n) {
    int i = blockIdx.x * blockDim.x + threadIdx.x;
    if (i < n) w[i] = 0u;
}

// -------------------------------------------------- per-(b,k) argmax over P
// prior_for_obj[b,k] = first index p maximizing IoU(box[b,k], prior[p])
__global__ void k_best_prior(const float* __restrict__ boxes,
                             const float* __restrict__ priors,   // cxcywh
                             int* __restrict__ pfo) {
    int bk = blockIdx.x;                 // 0 .. B_*K_-1
    const float* bx = boxes + (size_t)bk * 4;
    float ax1 = bx[0], ay1 = bx[1], ax2 = bx[2], ay2 = bx[3];
    float areaA = (ax2 - ax1) * (ay2 - ay1);

    float best = -1.0f; int bestp = 0;
    for (int p = threadIdx.x; p < P_; p += 256) {
        const float* pc = priors + (size_t)p * 4;
        float pcx = pc[0], pcy = pc[1], pw = pc[2], ph = pc[3];
        float px1 = pcx - 0.5f * pw, py1 = pcy - 0.5f * ph;
        float px2 = pcx + 0.5f * pw, py2 = pcy + 0.5f * ph;
        float lx = fmaxf(ax1, px1), ly = fmaxf(ay1, py1);
        float rx = fminf(ax2, px2), ry = fminf(ay2, py2);
        float w = fmaxf(rx - lx, 0.0f), h = fmaxf(ry - ly, 0.0f);
        float inter = w * h;
        float iou = inter / (areaA + pw * ph - inter);
        if (iou > best) { best = iou; bestp = p; }   // strict > => first max wins
    }
    __shared__ float sv[256];
    __shared__ int   si[256];
    sv[threadIdx.x] = best; si[threadIdx.x] = bestp;
    __syncthreads();
    for (int s = 128; s > 0; s >>= 1) {
        if ((int)threadIdx.x < s) {
            float ov = sv[threadIdx.x + s]; int oi = si[threadIdx.x + s];
            if (ov > sv[threadIdx.x] ||
                (ov == sv[threadIdx.x] && oi < si[threadIdx.x])) {
                sv[threadIdx.x] = ov; si[threadIdx.x] = oi;
            }
        }
        __syncthreads();
    }
    if (threadIdx.x == 0) pfo[bk] = si[0];
}

// ------------------------------- per-(b,p): match, encode, L1, CE, ce_neg row
__global__ void k_match_ce(const float* __restrict__ locs,
                           const float* __restrict__ scores,
                           const float* __restrict__ boxes,
                           const int*   __restrict__ labels,
                           const float* __restrict__ priors,
                           const int*   __restrict__ pfo,
                           float* __restrict__ ceneg,
                           int*   __restrict__ npos,
                           float* __restrict__ acc) {
    int b = blockIdx.y;
    int p = blockIdx.x * blockDim.x + threadIdx.x;

    __shared__ float sbox[K_ * 4];
    __shared__ int   slab[K_];
    __shared__ int   spfo[K_];
    if (threadIdx.x < K_ * 4) sbox[threadIdx.x] = boxes[(size_t)b * K_ * 4 + threadIdx.x];
    if (threadIdx.x < K_) {
        slab[threadIdx.x] = labels[b * K_ + threadIdx.x];
        spfo[threadIdx.x] = pfo[b * K_ + threadIdx.x];
    }
    __syncthreads();

    bool pos = false;
    float l1 = 0.0f, ce = 0.0f;

    if (p < P_) {
        const float* pc = priors + (size_t)p * 4;
        float pcx = pc[0], pcy = pc[1], pw = pc[2], ph = pc[3];
        float px1 = pcx - 0.5f * pw, py1 = pcy - 0.5f * ph;
        float px2 = pcx + 0.5f * pw, py2 = pcy + 0.5f * ph;
        float areaB = pw * ph;

        // argmax over K objects (first max wins)
        int obj = 0; float prmax = -1.0f;
        #pragma unroll
        for (int k = 0; k < K_; ++k) {
            float ax1 = sbox[k*4+0], ay1 = sbox[k*4+1];
            float ax2 = sbox[k*4+2], ay2 = sbox[k*4+3];
            float lx = fmaxf(ax1, px1), ly = fmaxf(ay1, py1);
            float rx = fminf(ax2, px2), ry = fminf(ay2, py2);
            float w = fmaxf(rx - lx, 0.0f), h = fmaxf(ry - ly, 0.0f);
            float inter = w * h;
            float iou = inter / ((ax2-ax1)*(ay2-ay1) + areaB - inter);
            if (iou > prmax) { prmax = iou; obj = k; }
        }
        // force each object to own its best prior (ascending k => last wins)
        #pragma unroll
        for (int k = 0; k < K_; ++k)
            if (spfo[k] == p) { obj = k; prmax = 0.6f; }   // THRESHOLD + 0.1

        int lbl = slab[obj];
        if (prmax < 0.5f) lbl = 0;
        pos = (lbl != 0);

        // encode(xy_to_cxcy(box[obj]), prior) and L1 vs predicted_locs
        float bx1 = sbox[obj*4+0], by1 = sbox[obj*4+1];
        float bx2 = sbox[obj*4+2], by2 = sbox[obj*4+3];
        float cx = 0.5f * (bx1 + bx2), cy = 0.5f * (by1 + by2);
        float bw = bx2 - bx1, bh = by2 - by1;
        float t0 = (cx - pcx) * 10.0f / pw;
        float t1 = (cy - pcy) * 10.0f / ph;
        float t2 = logf(bw / pw) * 5.0f;
        float t3 = logf(bh / ph) * 5.0f;
        const float* pl = locs + ((size_t)b * P_ + p) * 4;
        l1 = fabsf(pl[0]-t0) + fabsf(pl[1]-t1) + fabsf(pl[2]-t2) + fabsf(pl[3]-t3);

        // cross entropy: logsumexp(scores) - scores[lbl]
        const float* sc = scores + ((size_t)b * P_ + p) * C_;
        __builtin_prefetch(sc, 0, 0);                // global_prefetch_b8
        float mx = sc[0];
        for (int c = 1; c < C_; ++c) mx = fmaxf(mx, sc[c]);
        float sum = 0.0f;
        for (int c = 0; c < C_; ++c) sum += expf(sc[c] - mx);
        ce = mx + logf(sum) - sc[lbl];

        ceneg[(size_t)b * PPAD + p] = pos ? 0.0f : ce;
    }

    // wave32 shuffle reduction, one atomic triple per wave
    float l1c = pos ? l1 : 0.0f;
    float cec = pos ? ce : 0.0f;
    int   npc = pos ? 1 : 0;
    #pragma unroll
    for (int m = 16; m >= 1; m >>= 1) {
        l1c += __shfl_xor(l1c, m, 32);
        cec += __shfl_xor(cec, m, 32);
        npc += __shfl_xor(npc, m, 32);
    }
    if ((threadIdx.x & 31) == 0 && npc) {
        atomicAdd(&acc[0], l1c);
        atomicAdd(&acc[1], cec);
        atomicAdd(&npos[b], npc);
    }
}

// -------- per-image: async-copy row to LDS, bitonic sort desc, hard-neg sum
__global__ void __launch_bounds__(1024)
k_sort_hard(const float* __restrict__ ceneg,
            const int*   __restrict__ npos,
            float* __restrict__ acc) {
    int b = blockIdx.x;
    int tid = threadIdx.x;
    __shared__ __align__(16) float s[PPAD];          // 64 KB of the 320 KB/WGP
    const float* src = ceneg + (size_t)b * PPAD;

#if HAVE_ASYNC_LDS
    // CDNA5 async global->LDS (ASYNCcnt path), 16B per lane-op
    #pragma unroll
    for (int c = 0; c < 4; ++c) {
        int i = (tid + c * 1024) * 4;
        __builtin_amdgcn_global_load_async_to_lds_b128(
            (gv4i_ptr)(src + i),
            (lv4i_ptr)(s + i),
            0, 0);
    }
  #if __has_builtin(__builtin_amdgcn_s_wait_asynccnt)
    __builtin_amdgcn_s_wait_asynccnt(0);
  #else
    asm volatile("s_wait_asynccnt 0" ::: "memory");
  #endif
#else
    #pragma unroll
    for (int c = 0; c < 16; ++c) s[tid + c * 1024] = src[tid + c * 1024];
#endif
    __syncthreads();

    // bitonic sort, descending
    for (unsigned k = 2; k <= PPAD; k <<= 1) {
        for (unsigned j = k >> 1; j > 0; j >>= 1) {
            for (unsigned i = tid; i < PPAD; i += 1024) {
                unsigned ixj = i ^ j;
                if (ixj > i) {
                    float a = s[i], c2 = s[ixj];
                    bool desc = ((i & k) == 0);
                    if (desc ? (a < c2) : (a > c2)) { s[i] = c2; s[ixj] = a; }
                }
            }
            __syncthreads();
        }
    }

    // sum of ranks < min(3*n_pos, PPAD); zero-padding is sum-neutral (ce >= 0)
    int M = npos[b] * 3;
    if (M > PPAD) M = PPAD;
    float part = 0.0f;
    for (int i = tid; i < M; i += 1024) part += s[i];
    #pragma unroll
    for (int m = 16; m >= 1; m >>= 1) part += __shfl_xor(part, m, 32);
    __syncthreads();                                  // done reading s; reuse it
    if ((tid & 31) == 0) s[tid >> 5] = part;
    __syncthreads();
    if (tid < 32) {
        float v = s[tid];
        #pragma unroll
        for (int m = 16; m >= 1; m >>= 1) v += __shfl_xor(v, m, 32);
        if (tid == 0) atomicAdd(&acc[2], v);
    }
}

// ---------------------------------------------------------------- finalize
__global__ void k_final(const int* __restrict__ npos,
                        const float* __restrict__ acc,
                        float* __restrict__ out) {
    if (threadIdx.x == 0 && blockIdx.x == 0) {
        int t = 0;
        for (int b = 0; b < B_; ++b) t += npos[b];
        float npt = (float)t;
        out[0] = (acc[1] + acc[2]) / npt + acc[0] / (npt * 4.0f);
    }
}

extern "C" void kernel_launch(void* const* d_in, const int* in_sizes, int n_in,
                              void* d_out, int out_size, void* d_ws, size_t ws_size,
                              hipStream_t stream) {
    const float* locs   = (const float*)d_in[0];   // [B,P,4]
    const float* scores = (const float*)d_in[1];   // [B,P,C]
    const float* boxes  = (const float*)d_in[2];   // [B,K,4] xyxy
    const int*   labels = (const int*)  d_in[3];   // [B,K]
    const float* priors = (const float*)d_in[4];   // [P,4] cxcywh

    unsigned* ws    = (unsigned*)d_ws;
    float*    ceneg = (float*)(ws + OFF_CENEG);
    int*      npos  = (int*)  (ws + OFF_NPOS);
    int*      pfo   = (int*)  (ws + OFF_PFO);
    float*    acc   = (float*)(ws + OFF_ACC);
    float*    out   = (float*)d_out;

    k_zero<<<(WS_WORDS + 255) / 256, 256, 0, stream>>>(ws, WS_WORDS);

    k_best_prior<<<B_ * K_, 256, 0, stream>>>(boxes, priors, pfo);

    dim3 gridC((P_ + 255) / 256, B_);
    k_match_ce<<<gridC, 256, 0, stream>>>(locs, scores, boxes, labels, priors,
                                          pfo, ceneg, npos, acc);

    k_sort_hard<<<B_, 1024, 0, stream>>>(ceneg, npos, acc);

    k_final<<<1, 32, 0, stream>>>(npos, acc, out);
}